// BBoxHead_2559800508426
// MI455X (gfx1250) — compile-verified
//
#include <hip/hip_runtime.h>
#include <hip/hip_bf16.h>
#include <math.h>

typedef __attribute__((ext_vector_type(2))) float v2f;
typedef __attribute__((ext_vector_type(8))) float v8f;

#define BATCH   8
#define RROIS   1000
#define CDIM    1024
#define NC      81
#define NPAD    96          // 81 cls + 4 delta + 11 zero, multiple of 16
#define NROWS   (BATCH * RROIS)
#define MAXINST 100
#define NEGV    (-1e9f)
#define NMS_THR 0.5f
#define MIN_CONF 0.05f
#define NSORT   1024

// ---------------- pack w_cls | w_delta | zeros into W[1024,96] ----------------
__global__ void pack_w_kernel(const float* __restrict__ w_cls,
                              const float* __restrict__ w_delta,
                              float* __restrict__ Wp) {
    int i = blockIdx.x * blockDim.x + threadIdx.x;
    if (i >= CDIM * NPAD) return;
    int k = i / NPAD, n = i % NPAD;
    float v = 0.0f;
    if (n < NC)            v = w_cls[k * NC + n];
    else if (n < NC + 4)   v = w_delta[k * 4 + (n - NC)];
    Wp[i] = v;
}

// ---------------- fused GEMM: out[8000,96] = x[8000,1024] @ W[1024,96] --------
// One wave per 16x16 output tile, V_WMMA_F32_16X16X4_F32 over K.
__global__ __launch_bounds__(128) void gemm_wmma_kernel(
        const float* __restrict__ x,
        const float* __restrict__ Wp,
        float* __restrict__ out) {
    int wave = threadIdx.x >> 5;              // 0..3
    int lane = threadIdx.x & 31;
    int mtile = blockIdx.x * 4 + wave;        // 0..499
    int ntile = blockIdx.y;                   // 0..5
    int m0 = mtile * 16;
    int n0 = ntile * 16;

    int half  = lane >> 4;                    // 0: K {0,1}; 1: K {2,3}
    int sub   = lane & 15;
    int arow  = m0 + sub;                     // A: M = lane index
    int bcol  = n0 + sub;                     // B: N = lane index

    const float* __restrict__ xp = x  + (size_t)arow * CDIM + half * 2;
    const float* __restrict__ wp = Wp + (size_t)(half * 2) * NPAD + bcol;

    v8f acc = {};
    for (int k = 0; k < CDIM; k += 4) {
        if ((k & 63) == 0)
            __builtin_prefetch(xp + k + 64, 0, 0);   // global_prefetch_b8
        v2f a = *(const v2f*)(xp + k);               // K = k+2*half, k+2*half+1
        v2f b;
        b.x = wp[(size_t)k * NPAD];                  // W[k+2*half][col]
        b.y = wp[(size_t)(k + 1) * NPAD];            // W[k+2*half+1][col]
        acc = __builtin_amdgcn_wmma_f32_16x16x4_f32(
                false, a, false, b, (short)0, acc, false, false);
    }

    // C/D layout: VGPR v, lanes 0-15 -> M=v, lanes 16-31 -> M=v+8; N = lane&15
    int orow = m0 + (half << 3);
    int ocol = n0 + sub;
#pragma unroll
    for (int v = 0; v < 8; ++v)
        out[(size_t)(orow + v) * NPAD + ocol] = acc[v];
}

// ---------------- per-row: softmax/argmax, delta2bbox, clip -------------------
__global__ void postproc_kernel(const float* __restrict__ logits,
                                const float* __restrict__ b_cls,
                                const float* __restrict__ b_delta,
                                const float* __restrict__ rois,
                                float* __restrict__ boxesO,
                                float* __restrict__ scoresO,
                                int*   __restrict__ clsO) {
    int r = blockIdx.x * blockDim.x + threadIdx.x;
    if (r >= NROWS) return;
    const float* L = logits + (size_t)r * NPAD;

    float mx = -INFINITY; int am = 0;
    for (int j = 0; j < NC; ++j) {
        float l = L[j] + b_cls[j];
        if (l > mx) { mx = l; am = j; }     // first max, as jnp.argmax
    }
    float sum = 0.0f;
    for (int j = 0; j < NC; ++j)
        sum += expf(L[j] + b_cls[j] - mx);
    float score = 1.0f / sum;               // softmax at argmax

    float d0 = (L[NC + 0] + b_delta[0]) * 0.1f;
    float d1 = (L[NC + 1] + b_delta[1]) * 0.1f;
    float d2 = (L[NC + 2] + b_delta[2]) * 0.2f;
    float d3 = (L[NC + 3] + b_delta[3]) * 0.2f;

    const float* ro = rois + (size_t)r * 5;
    float y1 = ro[1] * 1024.0f, x1 = ro[2] * 1024.0f;
    float y2 = ro[3] * 1024.0f, x2 = ro[4] * 1024.0f;
    float h = y2 - y1, w = x2 - x1;
    float cy = y1 + 0.5f * h + d0 * h;
    float cx = x1 + 0.5f * w + d1 * w;
    h *= expf(d2);
    w *= expf(d3);
    float ny1 = fminf(fmaxf(cy - 0.5f * h, 0.0f), 1024.0f);
    float nx1 = fminf(fmaxf(cx - 0.5f * w, 0.0f), 1024.0f);
    float ny2 = fminf(fmaxf(cy + 0.5f * h, 0.0f), 1024.0f);
    float nx2 = fminf(fmaxf(cx + 0.5f * w, 0.0f), 1024.0f);

    boxesO[(size_t)r * 4 + 0] = ny1;
    boxesO[(size_t)r * 4 + 1] = nx1;
    boxesO[(size_t)r * 4 + 2] = ny2;
    boxesO[(size_t)r * 4 + 3] = nx2;
    bool valid = (am > 0) && (score >= MIN_CONF);
    scoresO[r] = valid ? score : NEGV;      // pre-masked like reference
    clsO[r] = am;
}

// ---------------- per-batch class-aware NMS (1 block / batch) -----------------
__global__ __launch_bounds__(NSORT) void nms_kernel(
        const float* __restrict__ boxes,
        const float* __restrict__ scores,
        const int*   __restrict__ cls,
        float* __restrict__ det) {
    __shared__ float s[NSORT];
    __shared__ int   id[NSORT];
    __shared__ float by1[NSORT], bx1[NSORT], by2[NSORT], bx2[NSORT];
    __shared__ int   sc[NSORT];
    __shared__ unsigned char supp[NSORT];
    __shared__ unsigned char keepA[NSORT];
    __shared__ int keepFlag;

    int b   = blockIdx.x;
    int tid = threadIdx.x;
    int base = b * RROIS;

    // zero this batch's detections (d_out is poisoned)
    for (int i = tid; i < MAXINST * 6; i += NSORT)
        det[b * MAXINST * 6 + i] = 0.0f;

    s[tid]  = (tid < RROIS) ? scores[base + tid] : NEGV;
    id[tid] = tid;
    __syncthreads();

    // bitonic sort, descending score with index tie-break (stable argsort)
    for (int k = 2; k <= NSORT; k <<= 1) {
        for (int j = k >> 1; j > 0; j >>= 1) {
            int ixj = tid ^ j;
            if (ixj > tid) {
                float s1 = s[tid], s2 = s[ixj];
                int   i1 = id[tid], i2 = id[ixj];
                bool before = (s1 > s2) || (s1 == s2 && i1 < i2);
                bool up = ((tid & k) == 0);
                if (before != up) {
                    s[tid] = s2; s[ixj] = s1;
                    id[tid] = i2; id[ixj] = i1;
                }
            }
            __syncthreads();
        }
    }

    // gather sorted boxes / classes
    int oi = id[tid];
    if (oi < RROIS) {
        int g = base + oi;
        by1[tid] = boxes[(size_t)g * 4 + 0];
        bx1[tid] = boxes[(size_t)g * 4 + 1];
        by2[tid] = boxes[(size_t)g * 4 + 2];
        bx2[tid] = boxes[(size_t)g * 4 + 3];
        sc[tid]  = cls[g];
    } else {
        by1[tid] = bx1[tid] = by2[tid] = bx2[tid] = 0.0f;
        sc[tid]  = -1;
    }
    supp[tid] = 0;
    keepA[tid] = 0;

    // sequential greedy suppression (matches reference scan semantics)
    for (int i = 0; i < RROIS; ++i) {
        __syncthreads();
        if (tid == 0) {
            int kf = (s[i] > 0.0f) && !supp[i];
            keepFlag = kf;
            keepA[i] = (unsigned char)kf;
        }
        __syncthreads();
        if (keepFlag && sc[tid] == sc[i]) {
            float yy1 = fmaxf(by1[tid], by1[i]);
            float xx1 = fmaxf(bx1[tid], bx1[i]);
            float yy2 = fminf(by2[tid], by2[i]);
            float xx2 = fminf(bx2[tid], bx2[i]);
            float inter = fmaxf(yy2 - yy1, 0.0f) * fmaxf(xx2 - xx1, 0.0f);
            float aT = (by2[tid] - by1[tid]) * (bx2[tid] - bx1[tid]);
            float aI = (by2[i] - by1[i]) * (bx2[i] - bx1[i]);
            float uni = fmaxf(aT + aI - inter, 1e-8f);
            if (inter / uni > NMS_THR) supp[tid] = 1;
        }
    }
    __syncthreads();

    // ordered compaction of kept detections (already score-descending)
    if (tid == 0) {
        int cnt = 0;
        for (int i = 0; i < RROIS && cnt < MAXINST; ++i) {
            if (keepA[i]) {
                float* drow = det + ((size_t)b * MAXINST + cnt) * 6;
                drow[0] = by1[i];
                drow[1] = bx1[i];
                drow[2] = by2[i];
                drow[3] = bx2[i];
                drow[4] = (float)sc[i];
                drow[5] = s[i];
                ++cnt;
            }
        }
    }
}

// ------------------------------------------------------------------------------
extern "C" void kernel_launch(void* const* d_in, const int* in_sizes, int n_in,
                              void* d_out, int out_size, void* d_ws, size_t ws_size,
                              hipStream_t stream) {
    const float* x       = (const float*)d_in[0];
    const float* w_cls   = (const float*)d_in[1];
    const float* b_cls   = (const float*)d_in[2];
    const float* w_delta = (const float*)d_in[3];
    const float* b_delta = (const float*)d_in[4];
    const float* rois    = (const float*)d_in[5];
    float* det = (float*)d_out;

    // workspace layout (floats): W[1024*96] | logits[8000*96] | boxes[8000*4] | scores[8000] | cls[8000]
    float* Wp      = (float*)d_ws;
    float* logits  = Wp + CDIM * NPAD;
    float* boxesW  = logits + (size_t)NROWS * NPAD;
    float* scoresW = boxesW + (size_t)NROWS * 4;
    int*   clsW    = (int*)(scoresW + NROWS);

    pack_w_kernel<<<(CDIM * NPAD + 255) / 256, 256, 0, stream>>>(w_cls, w_delta, Wp);

    dim3 ggrid(500 / 4, NPAD / 16);   // 125 x 6, 4 waves (tiles) per block
    gemm_wmma_kernel<<<ggrid, 128, 0, stream>>>(x, Wp, logits);

    postproc_kernel<<<(NROWS + 255) / 256, 256, 0, stream>>>(
        logits, b_cls, b_delta, rois, boxesW, scoresW, clsW);

    nms_kernel<<<BATCH, NSORT, 0, stream>>>(boxesW, scoresW, clsW, det);
}